// UniGINConv_81020263071814
// MI455X (gfx1250) — compile-verified
//
#include <hip/hip_runtime.h>

#define DCH 128         // feature dim D
#define M_EDGES 20000   // N_HEDGES (fixed by the reference)

typedef float v2f __attribute__((ext_vector_type(2)));
typedef float v8f __attribute__((ext_vector_type(8)));

// ---------------- zero workspace ----------------
__global__ void zero_ws_kernel(float* __restrict__ p, size_t n) {
  size_t i = (size_t)blockIdx.x * blockDim.x + threadIdx.x;
  if (i < n) p[i] = 0.0f;
}

// ---------------- pair -> hyperedge scatter-sum (+counts) ----------------
// one wave32 per pair; each lane moves float4 (32*4 = 128 floats = one row)
__global__ void scatter_edges_kernel(const float* __restrict__ X,
                                     const int* __restrict__ vertex,
                                     const int* __restrict__ edges,
                                     float* __restrict__ sums,
                                     float* __restrict__ counts, int P) {
  int wave = (int)(((size_t)blockIdx.x * blockDim.x + threadIdx.x) >> 5);
  int lane = threadIdx.x & 31;
  if (wave >= P) return;
  int v = vertex[wave];
  int e = edges[wave];
  float4 val = ((const float4*)(X + (size_t)v * DCH))[lane];
  float* dst = sums + (size_t)e * DCH + lane * 4;
  atomicAdd(dst + 0, val.x);
  atomicAdd(dst + 1, val.y);
  atomicAdd(dst + 2, val.z);
  atomicAdd(dst + 3, val.w);
  if (lane == 0) atomicAdd(counts + e, 1.0f);
}

// ---------------- Xe = sums / max(counts,1), in place ----------------
__global__ void normalize_edges_kernel(float* __restrict__ sums,
                                       const float* __restrict__ counts, size_t n) {
  size_t i = (size_t)blockIdx.x * blockDim.x + threadIdx.x;
  if (i >= n) return;
  float c = counts[i >> 7];          // i / DCH
  sums[i] = sums[i] / fmaxf(c, 1.0f);
}

// ---------------- hyperedge -> node scatter-sum ----------------
__global__ void scatter_nodes_kernel(const float* __restrict__ Xe,
                                     const int* __restrict__ vertex,
                                     const int* __restrict__ edges,
                                     float* __restrict__ xv, int P) {
  int wave = (int)(((size_t)blockIdx.x * blockDim.x + threadIdx.x) >> 5);
  int lane = threadIdx.x & 31;
  if (wave >= P) return;
  int v = vertex[wave];
  int e = edges[wave];
  float4 val = ((const float4*)(Xe + (size_t)e * DCH))[lane];
  float* dst = xv + (size_t)v * DCH + lane * 4;
  atomicAdd(dst + 0, val.x);
  atomicAdd(dst + 1, val.y);
  atomicAdd(dst + 2, val.z);
  atomicAdd(dst + 3, val.w);
}

// ---------------- Y = (1+eps)*X + Xv, in place over xv ----------------
__global__ void residual_kernel(const float* __restrict__ X,
                                const float* __restrict__ eps,
                                float* __restrict__ xv, size_t n) {
  size_t i = (size_t)blockIdx.x * blockDim.x + threadIdx.x;
  if (i >= n) return;
  xv[i] = fmaf(1.0f + eps[0], X[i], xv[i]);
}

// ---------------- out = Y @ W via V_WMMA_F32_16X16X4_F32 ----------------
// one wave per 16x16 output tile; K=128 in 32 steps of 4.
// A layout (ISA 7.12.2, 32-bit A 16x4): lane l holds row l%16, K = (l/16)*2 + {0,1}
// B layout (4x16, row striped across lanes): lane l holds col l%16, same K pair
// C/D layout: VGPR r holds row r + (l/16)*8, col l%16
__global__ void gemm_wmma_f32_kernel(const float* __restrict__ Y,
                                     const float* __restrict__ W,
                                     float* __restrict__ out, int numTiles) {
  int wave = (int)(((size_t)blockIdx.x * blockDim.x + threadIdx.x) >> 5);
  if (wave >= numTiles) return;            // wave-uniform; never splits EXEC
  int lane = threadIdx.x & 31;
  int colTile = wave & 7;                  // DCH/16 = 8 column tiles
  int rowTile = wave >> 3;
  int mn = lane & 15;
  int kh = (lane >> 4) << 1;               // 0 or 2

  const float* yrow = Y + (size_t)(rowTile * 16 + mn) * DCH;
  const float* wcol = W + (colTile * 16 + mn);

  v8f c = {};
#pragma unroll
  for (int kb = 0; kb < DCH; kb += 4) {
    v2f a, b;
    a.x = yrow[kb + kh];
    a.y = yrow[kb + kh + 1];
    b.x = wcol[(size_t)(kb + kh) * DCH];
    b.y = wcol[(size_t)(kb + kh + 1) * DCH];
    c = __builtin_amdgcn_wmma_f32_16x16x4_f32(false, a, false, b,
                                              (short)0, c, false, false);
  }

  float* o = out + (size_t)(rowTile * 16 + (lane >> 4) * 8) * DCH + colTile * 16 + mn;
#pragma unroll
  for (int r = 0; r < 8; ++r)
    o[(size_t)r * DCH] = c[r];
}

extern "C" void kernel_launch(void* const* d_in, const int* in_sizes, int n_in,
                              void* d_out, int out_size, void* d_ws, size_t ws_size,
                              hipStream_t stream) {
  const float* X      = (const float*)d_in[0];
  const float* W      = (const float*)d_in[1];
  const float* eps    = (const float*)d_in[2];
  const int*   vertex = (const int*)d_in[3];
  const int*   edges  = (const int*)d_in[4];

  const int N = in_sizes[0] / DCH;   // 100000
  const int P = in_sizes[3];         // 640000
  const int M = M_EDGES;             // 20000

  // workspace layout: sums[M*D] | counts[M (padded to 64)] | xv[N*D]
  float* sums   = (float*)d_ws;
  const size_t countsPad = ((size_t)M + 63) & ~(size_t)63;
  float* counts = sums + (size_t)M * DCH;
  float* xv     = counts + countsPad;
  const size_t zeroN = (size_t)M * DCH + countsPad + (size_t)N * DCH;

  zero_ws_kernel<<<(unsigned)((zeroN + 255) / 256), 256, 0, stream>>>((float*)d_ws, zeroN);

  // 8 waves (pairs) per 256-thread block
  scatter_edges_kernel<<<(unsigned)((P + 7) / 8), 256, 0, stream>>>(
      X, vertex, edges, sums, counts, P);

  const size_t en = (size_t)M * DCH;
  normalize_edges_kernel<<<(unsigned)((en + 255) / 256), 256, 0, stream>>>(sums, counts, en);

  scatter_nodes_kernel<<<(unsigned)((P + 7) / 8), 256, 0, stream>>>(
      sums, vertex, edges, xv, P);

  const size_t nn = (size_t)N * DCH;
  residual_kernel<<<(unsigned)((nn + 255) / 256), 256, 0, stream>>>(X, eps, xv, nn);

  const int numTiles = (N / 16) * (DCH / 16);  // 50000, exact
  gemm_wmma_f32_kernel<<<(unsigned)((numTiles + 7) / 8), 256, 0, stream>>>(
      xv, W, (float*)d_out, numTiles);
}